// MLPKANlayer_89584427860005
// MI455X (gfx1250) — compile-verified
//
#include <hip/hip_runtime.h>

typedef __attribute__((ext_vector_type(2))) float v2f;
typedef __attribute__((ext_vector_type(8))) float v8f;

#define INPUT_SIZE  128
#define OUTPUT_SIZE 128
#define BATCH       2048
#define XSTR        132   // LDS row stride (floats); 132 % 64 = 4 -> conflict-free broadcast reads
#define WAVES       8     // waves per workgroup, one j per wave

// sched_barrier mask: only SALU + VMEM + DS may cross. Blocks ALL(0x1),
// VALU(0x2) and MFMA/WMMA(0x8) so neither the WMMAs sink nor the dependent
// ReLU VALU/asm ops hoist across the phase boundary.
#define SCHED_BAR_PHASE 0x3F4

// Single-instruction ReLU. Inline asm so LLVM cannot refold it into
// canonicalize(v_max x,x) + v_max(0,x). v_max_num_f32 quiets NaNs in HW,
// so this is exactly IEEE maximumNumber(x, 0). VOP2: constant goes in src0.
static __device__ __forceinline__ float relu1(float v) {
    float r;
    asm("v_max_num_f32 %0, 0, %1" : "=v"(r) : "v"(v));
    return r;
}

// Fused KAN-style MLP bank: 16384 nets of 1->5->5->1, summed over the 128
// input-index nets per output column.
//
// Per wave: fixed j, 64 batch elements (4 WMMA column tiles of 16).
// Middle layer runs on V_WMMA_F32_16X16X4_F32, two chained K=4 steps over an
// augmented K=6 (bias folded in as a K-column against a row of ones in B).
// A sched_barrier keeps all 8 WMMAs of an iteration above the ReLU/layer-2
// phase so the 4 D tiles get distinct registers and pipeline in the matrix
// unit instead of serializing through one destination set.
__global__ __launch_bounds__(256)
void kan_fused_wmma(const float* __restrict__ x,
                    const float* __restrict__ w0, const float* __restrict__ b0,
                    const float* __restrict__ w1, const float* __restrict__ b1,
                    const float* __restrict__ w2, const float* __restrict__ b2,
                    float* __restrict__ out)
{
    __shared__ float xs[64 * XSTR];   // 33,792 B

    const int tid   = threadIdx.x;
    const int lane  = tid & 31;
    const int wave  = tid >> 5;
    const int m     = lane & 15;      // row (A/D) or batch column (B/D)
    const int h     = lane >> 4;      // lane half -> K offset of 2 in A/B layout
    const int bbase = blockIdx.x * 64;
    const int j     = blockIdx.y * WAVES + wave;

    // ---- stage x tile [64 batch x 128 i] into padded LDS, float4 coalesced ----
#pragma unroll
    for (int r = 0; r < 8; ++r) {
        int idx4 = r * 256 + tid;            // 2048 float4 total
        int b    = idx4 >> 5;                // 32 float4 per 128-float row
        int i4   = idx4 & 31;
        const float4 v =
            reinterpret_cast<const float4*>(x + (size_t)(bbase + b) * INPUT_SIZE)[i4];
        *reinterpret_cast<float4*>(&xs[b * XSTR + i4 * 4]) = v;
    }
    __syncthreads();

    float acc[4] = {0.f, 0.f, 0.f, 0.f};
    float b2sum  = 0.f;
    const bool  mok     = (m < 5);
    const int   mi      = mok ? m : 0;       // clamp so speculative loads stay in-bounds
    const float ones_hi = h ? 0.f : 1.0f;    // B_hi row K=5 (ones) / K=7 (zero), loop-invariant

    for (int i = 0; i < INPUT_SIZE; ++i) {
        const int n = (i << 7) + j;          // net index = i*OUTPUT_SIZE + j (wave-uniform)
        const float* __restrict__ w1n = w1 + (size_t)n * 25;
        const float* __restrict__ w0n = w0 + (size_t)n * 5;
        const float* __restrict__ b0n = b0 + (size_t)n * 5;
        const float* __restrict__ w2n = w2 + (size_t)n * 5;

        if (i + 1 < INPUT_SIZE)              // next net's W1 -> global_prefetch_b8
            __builtin_prefetch(w1 + (size_t)(n + OUTPUT_SIZE) * 25, 0, 3);

        // wave-uniform layer-0 / layer-2 weights (SMEM s_loads)
        const float w0c0 = w0n[0], w0c1 = w0n[1], w0c2 = w0n[2], w0c3 = w0n[3], w0c4 = w0n[4];
        const float b0c0 = b0n[0], b0c1 = b0n[1], b0c2 = b0n[2], b0c3 = b0n[3], b0c4 = b0n[4];
        const float w2c0 = w2n[0], w2c1 = w2n[1], w2c2 = w2n[2], w2c3 = w2n[3], w2c4 = w2n[4];
        b2sum += b2[n];

        // ---- A tiles (16x4 f32): lane m = row, VGPR v + 2h = K ----
        // A_lo: K=0..3 of W1aug; A_hi: K=4 -> w1 col4, K=5 -> b1, K=6,7 -> 0
        v2f alo, ahi;
        {
            float t0 = w1n[mi * 5 + 2 * h];         // K = 0 + 2h
            float t1 = w1n[mi * 5 + 1 + 2 * h];     // K = 1 + 2h
            float t2 = w1n[mi * 5 + 4];             // K = 4   (h==0 only)
            float t3 = b1[(size_t)n * 5 + mi];      // K = 5   (h==0 only)
            alo.x = mok ? t0 : 0.f;
            alo.y = mok ? t1 : 0.f;
            ahi.x = (mok && h == 0) ? t2 : 0.f;
            ahi.y = (mok && h == 0) ? t3 : 0.f;
        }

        // per-lane-half layer-0 weight selection for B rows (row K = v + 2h)
        const float w0s0 = h ? w0c2 : w0c0, b0s0 = h ? b0c2 : b0c0;
        const float w0s1 = h ? w0c3 : w0c1, b0s1 = h ? b0c3 : b0c1;

        // ---- phase 1: build B operands, issue all 8 WMMAs into 4 live D sets ----
        v8f d[4];
#pragma unroll
        for (int t = 0; t < 4; ++t) {
            const float xv = xs[(t * 16 + m) * XSTR + i];   // conflict-free broadcast

            // B tiles (4x16 f32): lane m = column (batch), VGPR v + 2h = row K
            v2f blo, bhi;
            blo.x = fmaxf(fmaf(w0s0, xv, b0s0), 0.f);           // K = 0 / 2
            blo.y = fmaxf(fmaf(w0s1, xv, b0s1), 0.f);           // K = 1 / 3
            bhi.x = h ? 0.f : fmaxf(fmaf(w0c4, xv, b0c4), 0.f); // K = 4 ; K=6 -> 0
            bhi.y = ones_hi;                                    // K = 5 ones ; K=7 -> 0

            // D = W1aug * H0aug  (rows 0..4 valid, rest exact zero)
            v8f dz = {};
            dz = __builtin_amdgcn_wmma_f32_16x16x4_f32(false, alo, false, blo,
                                                       (short)0, dz, false, false);
            d[t] = __builtin_amdgcn_wmma_f32_16x16x4_f32(false, ahi, false, bhi,
                                                         (short)0, dz, false, false);
        }

        // Hard phase boundary: WMMAs stay above, ReLU/FMA VALU stays below,
        // only SALU/VMEM/DS may cross. Forces 4 distinct live D tiles.
        __builtin_amdgcn_sched_barrier(SCHED_BAR_PHASE);

        // ---- phase 2: ReLU + layer-2 scalar-operand FMAs, accumulate over i ----
#pragma unroll
        for (int t = 0; t < 4; ++t) {
            float r;
            r = w2c0 * relu1(d[t][0]);
            r = fmaf(w2c1, relu1(d[t][1]), r);
            r = fmaf(w2c2, relu1(d[t][2]), r);
            r = fmaf(w2c3, relu1(d[t][3]), r);
            r = fmaf(w2c4, relu1(d[t][4]), r);
            acc[t] += r;
        }
    }

    // lanes 0-15 of each D tile hold batch columns 0-15; high lanes are duplicates/zeros
    if (lane < 16) {
#pragma unroll
        for (int t = 0; t < 4; ++t)
            out[(size_t)(bbase + t * 16 + m) * OUTPUT_SIZE + j] = acc[t] + b2sum;
    }
}

extern "C" void kernel_launch(void* const* d_in, const int* in_sizes, int n_in,
                              void* d_out, int out_size, void* d_ws, size_t ws_size,
                              hipStream_t stream) {
    (void)in_sizes; (void)n_in; (void)d_ws; (void)ws_size; (void)out_size;
    const float* x  = (const float*)d_in[0];
    const float* w0 = (const float*)d_in[1];
    const float* b0 = (const float*)d_in[2];
    const float* w1 = (const float*)d_in[3];
    const float* b1 = (const float*)d_in[4];
    const float* w2 = (const float*)d_in[5];
    const float* b2 = (const float*)d_in[6];

    dim3 grid(BATCH / 64, OUTPUT_SIZE / WAVES, 1);   // 32 x 16 = 512 workgroups
    kan_fused_wmma<<<grid, 256, 0, stream>>>(x, w0, b0, w1, b1, w2, b2, (float*)d_out);
}